// MangoFanbeamFpjLayer_43482248904785
// MI455X (gfx1250) — compile-verified
//
#include <hip/hip_runtime.h>

typedef __attribute__((ext_vector_type(2))) float v2f;
typedef __attribute__((ext_vector_type(8))) float v8f;

#define VIEWS 360
#define SGM   512
#define IMGD  512

// ---- geometry constants (match reference) ----
#define SID_C   750.0f
#define SDD_C   1250.0f
#define PIX_C   0.5f
#define ELE_C   1.2f
#define RFOV_C  181.01933598375618f   // 512*0.5*sqrt(2)/2
#define STEP_C  1.0f                  // FPJ_STEP * PIX
#define HALF_C  255.5f                // (IMG_DIM-1)/2
// N_STEPS = 363; we march 364 (91 * 4); the pad step provably samples 0.

// Fully branchless bilinear sample with zero-outside semantics.
// Clamped addresses -> 4 unconditional loads; validity handled by cndmask.
// EXEC is never modified, so the surrounding WMMA's all-ones EXEC holds.
__device__ __forceinline__ float bilin(const float* __restrict__ img,
                                       float col, float row) {
    float fc = floorf(col), fr = floorf(row);
    float wc = col - fc,    wr = row - fr;
    int ci = (int)fc, ri = (int)fr;
    int c0 = min(max(ci,     0), IMGD - 1);
    int c1 = min(max(ci + 1, 0), IMGD - 1);
    int r0 = min(max(ri,     0), IMGD - 1);
    int r1 = min(max(ri + 1, 0), IMGD - 1);
    const float* __restrict__ p0 = img + (r0 << 9);
    const float* __restrict__ p1 = img + (r1 << 9);
    float v00 = p0[c0];
    float v01 = p0[c1];
    float v10 = p1[c0];
    float v11 = p1[c1];
    bool c0v = (unsigned)ci       < (unsigned)IMGD;
    bool c1v = (unsigned)(ci + 1) < (unsigned)IMGD;
    bool r0v = (unsigned)ri       < (unsigned)IMGD;
    bool r1v = (unsigned)(ri + 1) < (unsigned)IMGD;
    v00 = (r0v & c0v) ? v00 : 0.0f;
    v01 = (r0v & c1v) ? v01 : 0.0f;
    v10 = (r1v & c0v) ? v10 : 0.0f;
    v11 = (r1v & c1v) ? v11 : 0.0f;
    float omwr = 1.0f - wr, omwc = 1.0f - wc;
    return v00 * omwr * omwc + v01 * omwr * wc
         + v10 * wr   * omwc + v11 * wr   * wc;
}

// One wave32 = 16 detectors x 4 ray-steps per WMMA (V_WMMA_F32_16X16X4_F32).
// A layout (16x4 f32): lanes 0-15 => M=0..15; VGPR0 holds K=0 (lanes 0-15) /
// K=2 (lanes 16-31); VGPR1 holds K=1 / K=3.  B = all-ones => D rows = step sums
// accumulated exactly in f32 by the matrix pipe (free cross-lane reduction).
__global__ __launch_bounds__(256)
void fpj_wmma_kernel(const float* __restrict__ x, float* __restrict__ out) {
    const int lane = threadIdx.x & 31;
    const int wave = threadIdx.x >> 5;
    const int tile = (blockIdx.x << 3) | wave;        // 0..31 (16 dets each)
    const int view = blockIdx.y;
    const int bv   = blockIdx.z * VIEWS + view;
    const int m    = lane & 15;                       // detector within tile
    const int khi  = lane >> 4;                       // step-half selector
    const int det  = (tile << 4) | m;

    // per-view geometry (uniform per wave except detector index)
    float beta = 6.283185307179586f * (float)view / 360.0f;
    float sb, cb;
    sincosf(beta, &sb, &cb);
    const float srcx = SID_C * cb;
    const float srcy = SID_C * sb;
    const float kdc  = 1.0f - SDD_C / SID_C;          // detector-center scale
    const float u    = ((float)det - 255.5f) * ELE_C; // tangential offset
    // ray direction = normalize(det_pos - src)
    float rx = srcx * kdc + u * (-sb) - srcx;
    float ry = srcy * kdc + u * cb    - srcy;
    float nrm = sqrtf(rx * rx + ry * ry);
    const float dx = rx / nrm, dy = ry / nrm;
    const float t0 = -(srcx * dx + srcy * dy);        // closest approach

    const float* __restrict__ img = x + blockIdx.z * (IMGD * IMGD);

    v8f acc = {0.0f, 0.0f, 0.0f, 0.0f, 0.0f, 0.0f, 0.0f, 0.0f};
    v2f bone;
    bone[0] = 1.0f;
    bone[1] = 1.0f;

    const float inv_pix = 1.0f / PIX_C;

    // 364 steps = 91 WMMA iterations; body is straight-line (no branches),
    // so EXEC stays all-ones and the loop software-pipelines cleanly.
    for (int s = 0; s < 364; s += 4) {
        const int k0 = s + (khi << 1);
        // match reference: ts = -R_FOV + (k + 0.5)*STEP ; t = t0 + ts
        float ta = t0 + (-RFOV_C + ((float)k0 + 0.5f) * STEP_C);
        float tb = t0 + (-RFOV_C + ((float)k0 + 1.5f) * STEP_C);
        float pxa = fmaf(ta, dx, srcx), pya = fmaf(ta, dy, srcy);
        float pxb = fmaf(tb, dx, srcx), pyb = fmaf(tb, dy, srcy);
        float a0 = bilin(img, pxa * inv_pix + HALF_C, HALF_C - pya * inv_pix);
        float a1 = bilin(img, pxb * inv_pix + HALF_C, HALF_C - pyb * inv_pix);
        v2f a;
        a[0] = a0;   // K = 2*khi
        a[1] = a1;   // K = 2*khi + 1
        // D = A(16x4) * ones(4x16) + C  -> every column of D = row sums
        acc = __builtin_amdgcn_wmma_f32_16x16x4_f32(
            false, a, false, bone, (short)0, acc, false, false);
    }

    // D layout: VGPR r, lanes 0-15 -> (M=r, N=lane); lanes 16-31 -> (M=8+r).
    // All N columns identical; lane 0 emits dets [0..7], lane 16 dets [8..15].
    if (m == 0) {
        float* __restrict__ o = out + bv * SGM + (tile << 4) + (khi << 3);
        o[0] = acc[0] * STEP_C;
        o[1] = acc[1] * STEP_C;
        o[2] = acc[2] * STEP_C;
        o[3] = acc[3] * STEP_C;
        o[4] = acc[4] * STEP_C;
        o[5] = acc[5] * STEP_C;
        o[6] = acc[6] * STEP_C;
        o[7] = acc[7] * STEP_C;
    }
}

extern "C" void kernel_launch(void* const* d_in, const int* in_sizes, int n_in,
                              void* d_out, int out_size, void* d_ws, size_t ws_size,
                              hipStream_t stream) {
    (void)n_in; (void)d_ws; (void)ws_size; (void)out_size;
    const float* x = (const float*)d_in[0];
    float* out = (float*)d_out;
    const int B = in_sizes[0] / (IMGD * IMGD);        // batch (2)
    // grid: x = 4 blocks of 8 waves (32 tiles = 512 dets), y = views, z = batch
    dim3 grid(4, VIEWS, B), block(256);
    hipLaunchKernelGGL(fpj_wmma_kernel, grid, block, 0, stream, x, out);
}